// DSSIM_Loss_64226940944563
// MI455X (gfx1250) — compile-verified
//
#include <hip/hip_runtime.h>

typedef float v2f __attribute__((ext_vector_type(2)));
typedef float v4f __attribute__((ext_vector_type(4)));
typedef float v8f __attribute__((ext_vector_type(8)));

#define G_H 512
#define G_W 512
#define N_PLANES 48                      // 16 * 3
#define N_TILES (N_PLANES * 32 * 32)     // 49152 16x16 output tiles
#define WAVES_PER_BLOCK 8
#define N_BLOCKS (N_TILES / WAVES_PER_BLOCK)  // 6144; 128 blocks per plane
#define CSTR 36  // LDS column stride (floats): b128-store/b64-load friendly,
                 // conflict-free for the V-pass b64 reads

// 11-tap Gaussian, sigma=1.5, normalized (matches reference float64->float32)
__constant__ float c_g[11] = {
    0.00102842f, 0.00759875f, 0.03600077f, 0.10936069f, 0.21300553f,
    0.26601172f, 0.21300553f, 0.10936069f, 0.03600077f, 0.00759875f,
    0.00102842f};

__device__ __forceinline__ float gwin(int i) {
  return (i >= 0 && i < 11) ? c_g[i] : 0.0f;
}

__device__ __forceinline__ v8f wmma4(v2f a, v2f b, v8f c) {
  // D = A(16x4 f32) * B(4x16 f32) + C(16x16 f32)
  return __builtin_amdgcn_wmma_f32_16x16x4_f32(false, a, false, b, (short)0, c,
                                               false, false);
}

// H-pass: 5 signals (x, y, x^2, y^2, xy) x 2 row blocks of Toeplitz WMMA.
// xb/yb are block-uniform (SGPR base); all per-lane addressing is a 32-bit
// offset so loads lower to saddr+voffset form with immediate column offsets.
// EDGE=false: tile fully interior, unchecked.
// EDGE=true : clamp offsets in-bounds, load unconditionally, then select 0
//             for logically out-of-range taps (branchless; EXEC stays all-1s).
template <bool EDGE>
__device__ __forceinline__ void hpass(const float* __restrict__ xb,
                                      const float* __restrict__ yb, int r0,
                                      int c0, int h, int m,
                                      const float wt[7][2], v8f hacc[5][2]) {
#pragma unroll
  for (int b = 0; b < 2; ++b) {
    const int gr = r0 - 5 + 16 * b + m;  // image row for this lane
    if (!EDGE) {
      const int off = gr * G_W + (c0 - 5 + 2 * h);  // 32-bit lane offset
#pragma unroll
      for (int kk = 0; kk < 7; ++kk) {
        const float x0 = xb[off + 4 * kk], x1 = xb[off + 4 * kk + 1];
        const float y0 = yb[off + 4 * kk], y1 = yb[off + 4 * kk + 1];
        const v2f bw = {wt[kk][0], wt[kk][1]};
        v2f a;
        a = (v2f){x0, x1};           hacc[0][b] = wmma4(a, bw, hacc[0][b]);
        a = (v2f){y0, y1};           hacc[1][b] = wmma4(a, bw, hacc[1][b]);
        a = (v2f){x0 * x0, x1 * x1}; hacc[2][b] = wmma4(a, bw, hacc[2][b]);
        a = (v2f){y0 * y0, y1 * y1}; hacc[3][b] = wmma4(a, bw, hacc[3][b]);
        a = (v2f){x0 * y0, x1 * y1}; hacc[4][b] = wmma4(a, bw, hacc[4][b]);
      }
    } else {
      const bool rok = (unsigned)gr < (unsigned)G_H;
      const int rowoff = min(max(gr, 0), G_H - 1) * G_W;
#pragma unroll
      for (int kk = 0; kk < 7; ++kk) {
        const int gc = c0 - 5 + 4 * kk + 2 * h;
        const bool ok0 = rok && ((unsigned)gc < (unsigned)G_W);
        const bool ok1 = rok && ((unsigned)(gc + 1) < (unsigned)G_W);
        const int o0 = rowoff + min(max(gc, 0), G_W - 1);
        const int o1 = rowoff + min(max(gc + 1, 0), G_W - 1);
        float x0 = xb[o0], x1 = xb[o1], y0 = yb[o0], y1 = yb[o1];
        x0 = ok0 ? x0 : 0.0f;
        x1 = ok1 ? x1 : 0.0f;
        y0 = ok0 ? y0 : 0.0f;
        y1 = ok1 ? y1 : 0.0f;
        const v2f bw = {wt[kk][0], wt[kk][1]};
        v2f a;
        a = (v2f){x0, x1};           hacc[0][b] = wmma4(a, bw, hacc[0][b]);
        a = (v2f){y0, y1};           hacc[1][b] = wmma4(a, bw, hacc[1][b]);
        a = (v2f){x0 * x0, x1 * x1}; hacc[2][b] = wmma4(a, bw, hacc[2][b]);
        a = (v2f){y0 * y0, y1 * y1}; hacc[3][b] = wmma4(a, bw, hacc[3][b]);
        a = (v2f){x0 * y0, x1 * y1}; hacc[4][b] = wmma4(a, bw, hacc[4][b]);
      }
    }
  }
}

// One wave computes one 16x16 tile of all five depthwise convolutions via
// Toeplitz-matmul WMMA (H-pass then V-pass), then the SSIM map and a partial
// sum. 8 waves / block, per-wave private LDS scratch (column-major tile).
__global__ __launch_bounds__(256) void dssim_main(const float* __restrict__ x,
                                                  const float* __restrict__ y,
                                                  float* __restrict__ partial) {
  __shared__ __align__(16) float lds[WAVES_PER_BLOCK][16 * CSTR];
  __shared__ float blocksum[WAVES_PER_BLOCK];

  const int lane = threadIdx.x & 31;
  const int wave = threadIdx.x >> 5;
  const int h = lane >> 4;  // half-wave: selects K pair
  const int m = lane & 15;  // M for A-operand / N for B-operand

  // Per-lane Toeplitz weights: wt[kk][i] = g[(4kk + i + 2h) - m] (0 outside).
  // Serves as BOTH the H-pass B operand and the V-pass A operand (layout
  // symmetry of WMMA A/B striping).
  float wt[7][2];
#pragma unroll
  for (int kk = 0; kk < 7; ++kk)
#pragma unroll
    for (int i = 0; i < 2; ++i)
      wt[kk][i] = gwin(4 * kk + i + 2 * h - m);

  // plane is block-uniform (1024 tiles/plane, 8 waves/block) -> SGPR base.
  const int plane = blockIdx.x >> 7;                   // 128 blocks per plane
  const int rem = ((blockIdx.x & 127) << 3) | wave;    // tile index in plane
  const int r0 = (rem >> 5) << 4;
  const int c0 = (rem & 31) << 4;
  const float* xb = x + (size_t)plane * (G_H * G_W);
  const float* yb = y + (size_t)plane * (G_H * G_W);

  v8f hacc[5][2];
#pragma unroll
  for (int s = 0; s < 5; ++s)
#pragma unroll
    for (int b = 0; b < 2; ++b)
      hacc[s][b] = (v8f){0.f, 0.f, 0.f, 0.f, 0.f, 0.f, 0.f, 0.f};

  // Wave-uniform interior test: rows r0-5..r0+26 and cols c0-5..c0+22 all
  // in-bounds. 87.9% of tiles take the unchecked path.
  const bool interior =
      (r0 >= 5) && (r0 + 26 <= G_H - 1) && (c0 >= 5) && (c0 + 22 <= G_W - 1);
  if (interior)
    hpass<false>(xb, yb, r0, c0, h, m, wt, hacc);
  else
    hpass<true>(xb, yb, r0, c0, h, m, wt, hacc);

  // V-pass: bounce tmp through per-wave LDS, column-major so each lane's 8
  // D-rows are contiguous (2x ds_store_b128) and each B pair is one
  // ds_load_b64. Tmp rows >= 26 get weight 0 automatically.
  float* Lb = lds[wave];
  v8f vacc[5];
#pragma unroll
  for (int s = 0; s < 5; ++s) {
#pragma unroll
    for (int b = 0; b < 2; ++b) {
      const int rbase = 16 * b + 8 * h;  // D-matrix row base for this lane
      const v8f d = hacc[s][b];
      const v4f lo = {d[0], d[1], d[2], d[3]};
      const v4f hi = {d[4], d[5], d[6], d[7]};
      *(v4f*)(Lb + m * CSTR + rbase) = lo;
      *(v4f*)(Lb + m * CSTR + rbase + 4) = hi;
    }
    asm volatile("s_wait_dscnt 0x0" ::: "memory");  // cross-lane hand-off
    v8f acc = {0.f, 0.f, 0.f, 0.f, 0.f, 0.f, 0.f, 0.f};
#pragma unroll
    for (int kk = 0; kk < 7; ++kk) {
      const int k0 = 4 * kk + 2 * h;
      const v2f bv = *(const v2f*)(Lb + m * CSTR + k0);
      const v2f av = {wt[kk][0], wt[kk][1]};
      acc = wmma4(av, bv, acc);
    }
    vacc[s] = acc;
  }

  // SSIM map for this lane's 8 output pixels, then partial sums.
  const float C1v = 1.0e-4f;  // 0.01^2
  const float C2v = 9.0e-4f;  // 0.03^2
  float ssum = 0.0f;
#pragma unroll
  for (int v = 0; v < 8; ++v) {
    const float mu1 = vacc[0][v], mu2 = vacc[1][v];
    const float exx = vacc[2][v], eyy = vacc[3][v], exy = vacc[4][v];
    const float mu1s = mu1 * mu1, mu2s = mu2 * mu2, m12 = mu1 * mu2;
    const float s1 = exx - mu1s, s2 = eyy - mu2s, s12 = exy - m12;
    const float num = (2.0f * m12 + C1v) * (2.0f * s12 + C2v);
    const float den = (mu1s + mu2s + C1v) * (s1 + s2 + C2v);
    ssum += num / den;
  }
#pragma unroll
  for (int off = 16; off > 0; off >>= 1)
    ssum += __shfl_xor(ssum, off, 32);
  if (lane == 0) blocksum[wave] = ssum;
  __syncthreads();
  if (threadIdx.x == 0) {
    float t = 0.0f;
#pragma unroll
    for (int w = 0; w < WAVES_PER_BLOCK; ++w) t += blocksum[w];
    partial[blockIdx.x] = t;
  }
}

// Deterministic fixed-order reduction of the 6144 block partials.
__global__ __launch_bounds__(256) void dssim_finalize(
    const float* __restrict__ partial, float* __restrict__ out) {
  __shared__ double sh[256];
  double s = 0.0;
  for (int i = threadIdx.x; i < N_BLOCKS; i += 256) s += (double)partial[i];
  sh[threadIdx.x] = s;
  __syncthreads();
  for (int st = 128; st > 0; st >>= 1) {
    if ((int)threadIdx.x < st) sh[threadIdx.x] += sh[threadIdx.x + st];
    __syncthreads();
  }
  if (threadIdx.x == 0)
    out[0] = (float)(1.0 - sh[0] / 12582912.0);  // 16*3*512*512
}

extern "C" void kernel_launch(void* const* d_in, const int* in_sizes, int n_in,
                              void* d_out, int out_size, void* d_ws,
                              size_t ws_size, hipStream_t stream) {
  const float* x = (const float*)d_in[0];
  const float* y = (const float*)d_in[1];
  float* partial = (float*)d_ws;  // 6144 floats
  dssim_main<<<N_BLOCKS, 256, 0, stream>>>(x, y, partial);
  dssim_finalize<<<1, 256, 0, stream>>>(partial, (float*)d_out);
}